// ResGATv3_56564719288601
// MI455X (gfx1250) — compile-verified
//
#include <hip/hip_runtime.h>
#include <math.h>

// ResGATv2 forward for MI455X (gfx1250, wave32).
// fp32 end-to-end; dense GEMMs via V_WMMA_F32_16X16X4_F32 with async
// global->LDS A-tile staging (GLOBAL_LOAD_ASYNC_TO_LDS_B128);
// edge phase is wave-per-edge with coalesced loads and native f32 atomics.

#define HID   64
#define HEADS 8
#define FDIM  512   // HEADS*HID

typedef __attribute__((ext_vector_type(2))) float v2f;
typedef __attribute__((ext_vector_type(8))) float v8f;
typedef __attribute__((ext_vector_type(4))) int   v4i;

#if defined(__gfx1250__) && __has_builtin(__builtin_amdgcn_global_load_async_to_lds_b128)
#define HAVE_ASYNC_LDS 1
#else
#define HAVE_ASYNC_LDS 0
#endif

__device__ __forceinline__ void wait_async_lds() {
#if HAVE_ASYNC_LDS
#if __has_builtin(__builtin_amdgcn_s_wait_asynccnt)
  __builtin_amdgcn_s_wait_asynccnt(0);
#else
  asm volatile("s_wait_asynccnt 0" ::: "memory");
#endif
#endif
}

__device__ __forceinline__ float lrelu(float v, float s) { return v > 0.f ? v : v * s; }

// Sign-aware float atomic max (locations initialized to -inf).
__device__ __forceinline__ void atomicMaxF(float* addr, float v) {
  if (v >= 0.f) atomicMax((int*)addr, __float_as_int(v));
  else          atomicMin((unsigned int*)addr, __float_as_uint(v));
}

// ---------------------------------------------------------------------------
// C[M,N] = act(A[M,K] @ W[K,N] + bias[N])  via V_WMMA_F32_16X16X4_F32.
// Block = 128 threads = 4 waves; block covers 16 rows x 64 cols (1 tile/wave).
// Grid = (M/16 tiles, N/64). A tile staged in LDS, stride K+4 -> (K+4)%64==4
// gives conflict-free paired (b64) A-fragment reads for both lane halves.
// A frag (16x4 f32): lanes 0-15 hold {K=k,k+1}, lanes 16-31 hold {K=k+2,k+3}.
// B frag (4x16 f32): lanes 0-15 = rows k,k+1 at col=lane; lanes 16-31 = k+2,k+3.
// C/D: VGPR i -> M = i + (lane>=16 ? 8 : 0), N = lane&15.
// ---------------------------------------------------------------------------
template <int K, int N>
__global__ void gemm_wmma_f32(const float* __restrict__ A, const float* __restrict__ W,
                              const float* __restrict__ bias, float* __restrict__ C,
                              int M, int act) {
  constexpr int LDA = K + 4;
  __shared__ __align__(16) float As[16 * LDA];
  const int m0 = blockIdx.x * 16;
  const bool full = (m0 + 16 <= M);

  if (full) {
    // 16*K floats in 16B chunks; row starts are 16B aligned (K%4==0, pad 4).
    constexpr int CPR = K / 4;            // float4 chunks per row
    for (int i = threadIdx.x; i < 16 * CPR; i += blockDim.x) {
      int r = i / CPR, c4 = (i % CPR) * 4;
#if HAVE_ASYNC_LDS
      __builtin_amdgcn_global_load_async_to_lds_b128(
          (__attribute__((address_space(1))) v4i*)(A + (size_t)(m0 + r) * K + c4),
          (__attribute__((address_space(3))) v4i*)&As[r * LDA + c4], 0, 0);
#else
      *(float4*)&As[r * LDA + c4] = *(const float4*)(A + (size_t)(m0 + r) * K + c4);
#endif
    }
    wait_async_lds();
  } else {
    for (int i = threadIdx.x; i < 16 * K; i += blockDim.x) {
      int r = i / K, c = i % K;
      int gr = m0 + r;
      As[r * LDA + c] = (gr < M) ? A[(size_t)gr * K + c] : 0.f;
    }
  }
  __syncthreads();

  const int lane = threadIdx.x & 31;
  const int wv   = threadIdx.x >> 5;      // 4 col-tiles of 16
  const int mr   = lane & 15;
  const int kh   = (lane >> 4) << 1;      // 0 (lanes 0-15) or 2 (lanes 16-31)
  const int col  = blockIdx.y * 64 + wv * 16 + mr;

  v8f acc = {};
  #pragma unroll 8
  for (int k0 = 0; k0 < K; k0 += 4) {
    v2f a, b;
    a.x = As[mr * LDA + k0 + kh];
    a.y = As[mr * LDA + k0 + kh + 1];
    b.x = W[(size_t)(k0 + kh) * N + col];
    b.y = W[(size_t)(k0 + kh + 1) * N + col];
    acc = __builtin_amdgcn_wmma_f32_16x16x4_f32(false, a, false, b, (short)0, acc,
                                                false, false);
  }

  const float bv = bias[col];
  const int r0 = m0 + ((lane >> 4) << 3);
  if (full) {
    #pragma unroll
    for (int i = 0; i < 8; ++i) {
      float v = acc[i] + bv;
      if (act) v = lrelu(v, 0.01f);
      C[(size_t)(r0 + i) * N + col] = v;
    }
  } else {
    #pragma unroll
    for (int i = 0; i < 8; ++i) {
      int r = r0 + i;
      if (r < M) {
        float v = acc[i] + bv;
        if (act) v = lrelu(v, 0.01f);
        C[(size_t)r * N + col] = v;
      }
    }
  }
}

// agg[n][j] = gat_bias[j]; emax = -inf; denom = 0
__global__ void gat_init(float* __restrict__ agg, const float* __restrict__ gbias,
                         float* __restrict__ emax, float* __restrict__ denom, int n) {
  int i = blockIdx.x * blockDim.x + threadIdx.x;
  if (i < n * FDIM) agg[i] = gbias[i & (FDIM - 1)];
  if (i < n * HEADS) { emax[i] = -__builtin_inff(); denom[i] = 0.f; }
}

// One wave32 per edge: logits e[h] = sum_c att[h,c]*leaky(xl[s,h,c]+xr[d,h,c], 0.2)
// + running atomic segment-max over dst.
__global__ void edge_logits(const float* __restrict__ xl, const float* __restrict__ xr,
                            const float* __restrict__ att,
                            const int* __restrict__ src, const int* __restrict__ dst,
                            int nE, int nN, float* __restrict__ elog,
                            float* __restrict__ emax) {
  int e    = (blockIdx.x * blockDim.x + threadIdx.x) >> 5;
  int lane = threadIdx.x & 31;
  int tot  = nE + nN;
  if (e >= tot) return;
  int s, d;
  if (e < nE) { s = src[e]; d = dst[e]; } else { s = d = e - nE; }   // self-loops
  const float* L = xl + (size_t)s * FDIM;
  const float* R = xr + (size_t)d * FDIM;
  #pragma unroll
  for (int h = 0; h < HEADS; ++h) {
    int c0 = h * HID + lane;
    float v0 = lrelu(L[c0] + R[c0], 0.2f);
    float v1 = lrelu(L[c0 + 32] + R[c0 + 32], 0.2f);
    float p  = v0 * att[c0] + v1 * att[c0 + 32];
    #pragma unroll
    for (int o = 16; o > 0; o >>= 1) p += __shfl_xor(p, o, 32);
    if (lane == 0) {
      elog[(size_t)e * HEADS + h] = p;
      atomicMaxF(&emax[d * HEADS + h], p);
    }
  }
}

// ex = exp(e - emax[dst]); denom[dst] += ex   (thread per (edge,head))
__global__ void edge_exp(float* __restrict__ elog, const float* __restrict__ emax,
                         float* __restrict__ denom, const int* __restrict__ dst,
                         int nE, int nN) {
  int i = blockIdx.x * blockDim.x + threadIdx.x;
  if (i >= (nE + nN) * HEADS) return;
  int e = i >> 3, h = i & 7;
  int d = (e < nE) ? dst[e] : (e - nE);
  float ex = expf(elog[i] - emax[d * HEADS + h]);
  elog[i] = ex;
  atomicAdd(&denom[d * HEADS + h], ex);
}

// One wave per edge: agg[dst] += (ex/denom[dst]) * xl[src]; coalesced f32 atomics.
__global__ void edge_aggregate(const float* __restrict__ ex, const float* __restrict__ denom,
                               const float* __restrict__ xl, float* __restrict__ agg,
                               const int* __restrict__ src, const int* __restrict__ dst,
                               int nE, int nN) {
  int e    = (blockIdx.x * blockDim.x + threadIdx.x) >> 5;
  int lane = threadIdx.x & 31;
  int tot  = nE + nN;
  if (e >= tot) return;
  int s, d;
  if (e < nE) { s = src[e]; d = dst[e]; } else { s = d = e - nE; }
  const float* L = xl + (size_t)s * FDIM;
  float* O = agg + (size_t)d * FDIM;
  #pragma unroll
  for (int h = 0; h < HEADS; ++h) {
    float alpha = ex[(size_t)e * HEADS + h] / denom[d * HEADS + h];
    int c = h * HID + lane;
    atomicAdd(&O[c],      alpha * L[c]);
    atomicAdd(&O[c + 32], alpha * L[c + 32]);
  }
}

// out = LayerNorm(a + res) * g + b    (one wave per node, butterfly reductions)
__global__ void add_layernorm(const float* __restrict__ a, const float* __restrict__ res,
                              const float* __restrict__ g, const float* __restrict__ b,
                              float* __restrict__ out, int n) {
  int node = (blockIdx.x * blockDim.x + threadIdx.x) >> 5;
  int lane = threadIdx.x & 31;
  if (node >= n) return;
  size_t base = (size_t)node * HID;
  float x0 = a[base + lane]      + res[base + lane];
  float x1 = a[base + lane + 32] + res[base + lane + 32];
  float s = x0 + x1, q = x0 * x0 + x1 * x1;
  #pragma unroll
  for (int o = 16; o > 0; o >>= 1) { s += __shfl_xor(s, o, 32); q += __shfl_xor(q, o, 32); }
  float mean = s * (1.f / 64.f);
  float var  = q * (1.f / 64.f) - mean * mean;
  float rstd = rsqrtf(var + 1e-5f);
  out[base + lane]      = (x0 - mean) * rstd * g[lane]      + b[lane];
  out[base + lane + 32] = (x1 - mean) * rstd * g[lane + 32] + b[lane + 32];
}

// ---------------------------------------------------------------------------
// Host orchestration. Input order (flattened setup_inputs dict):
//   0:x 1:edge_index 2:w1 3:b1 4:w2 5:b2, then per layer (16 each):
//   lin_l_w, lin_l_b, lin_r_w, lin_r_b, att, gat_bias, post_w, post_b,
//   fc1_w, fc1_b, fc2_w, fc2_b, gatnorm_g, gatnorm_b, fcnorm_g, fcnorm_b
// ---------------------------------------------------------------------------
extern "C" void kernel_launch(void* const* d_in, const int* in_sizes, int n_in,
                              void* d_out, int out_size, void* d_ws, size_t ws_size,
                              hipStream_t stream) {
  const float* x  = (const float*)d_in[0];
  const int*   ei = (const int*)d_in[1];
  const int nE  = in_sizes[1] / 2;
  const int nN  = in_sizes[0] / HID;
  const int* src = ei;
  const int* dst = ei + nE;
  const int totE = nE + nN;

  const float* w1 = (const float*)d_in[2];
  const float* b1 = (const float*)d_in[3];
  const float* w2 = (const float*)d_in[4];
  const float* b2 = (const float*)d_in[5];

  // Workspace layout (~75 MB of f32)
  float* ws    = (float*)d_ws;
  float* h     = ws;  ws += (size_t)nN * HID;
  float* t0    = ws;  ws += (size_t)nN * HID;
  float* t1    = ws;  ws += (size_t)nN * HID;
  float* xl    = ws;  ws += (size_t)nN * FDIM;
  float* xr    = ws;  ws += (size_t)nN * FDIM;
  float* agg   = ws;  ws += (size_t)nN * FDIM;
  float* elog  = ws;  ws += (size_t)totE * HEADS;
  float* emax  = ws;  ws += (size_t)nN * HEADS;
  float* denom = ws;

  const int gm = (nN + 15) / 16;                  // 625 row tiles
  const dim3 blk(128);
  const int edgeBlocks = (totE + 7) / 8;          // 8 waves / 256-thread block
  const int lnBlocks   = (nN + 7) / 8;

  // Embed: Linear->LeakyReLU(0.01)->Linear
  gemm_wmma_f32<64, 64><<<dim3(gm, 1), blk, 0, stream>>>(x,  w1, b1, t0, nN, 1);
  gemm_wmma_f32<64, 64><<<dim3(gm, 1), blk, 0, stream>>>(t0, w2, b2, h,  nN, 0);

  for (int l = 0; l < 2; ++l) {
    const int p = 6 + 16 * l;
    const float* lin_l_w = (const float*)d_in[p + 0];
    const float* lin_l_b = (const float*)d_in[p + 1];
    const float* lin_r_w = (const float*)d_in[p + 2];
    const float* lin_r_b = (const float*)d_in[p + 3];
    const float* att     = (const float*)d_in[p + 4];
    const float* gbias   = (const float*)d_in[p + 5];
    const float* post_w  = (const float*)d_in[p + 6];
    const float* post_b  = (const float*)d_in[p + 7];
    const float* fc1_w   = (const float*)d_in[p + 8];
    const float* fc1_b   = (const float*)d_in[p + 9];
    const float* fc2_w   = (const float*)d_in[p + 10];
    const float* fc2_b   = (const float*)d_in[p + 11];
    const float* gn_g    = (const float*)d_in[p + 12];
    const float* gn_b    = (const float*)d_in[p + 13];
    const float* fn_g    = (const float*)d_in[p + 14];
    const float* fn_b    = (const float*)d_in[p + 15];

    gemm_wmma_f32<64, 512><<<dim3(gm, 8), blk, 0, stream>>>(h, lin_l_w, lin_l_b, xl, nN, 0);
    gemm_wmma_f32<64, 512><<<dim3(gm, 8), blk, 0, stream>>>(h, lin_r_w, lin_r_b, xr, nN, 0);

    gat_init<<<((size_t)nN * FDIM + 255) / 256, 256, 0, stream>>>(agg, gbias, emax, denom, nN);
    edge_logits   <<<edgeBlocks, 256, 0, stream>>>(xl, xr, att, src, dst, nE, nN, elog, emax);
    edge_exp      <<<(totE * HEADS + 255) / 256, 256, 0, stream>>>(elog, emax, denom, dst, nE, nN);
    edge_aggregate<<<edgeBlocks, 256, 0, stream>>>(elog, denom, xl, agg, src, dst, nE, nN);

    gemm_wmma_f32<512, 64><<<dim3(gm, 1), blk, 0, stream>>>(agg, post_w, post_b, t0, nN, 0);
    add_layernorm<<<lnBlocks, 256, 0, stream>>>(t0, h, gn_g, gn_b, h, nN);

    gemm_wmma_f32<64, 64><<<dim3(gm, 1), blk, 0, stream>>>(h,  fc1_w, fc1_b, t0, nN, 1);
    gemm_wmma_f32<64, 64><<<dim3(gm, 1), blk, 0, stream>>>(t0, fc2_w, fc2_b, t1, nN, 0);
    float* outp = (l == 1) ? (float*)d_out : h;
    add_layernorm<<<lnBlocks, 256, 0, stream>>>(t1, h, fn_g, fn_b, outp, nN);
  }
}